// KNNGroup_42099269435598
// MI455X (gfx1250) — compile-verified
//
#include <hip/hip_runtime.h>

typedef __attribute__((ext_vector_type(2))) float v2f;
typedef __attribute__((ext_vector_type(4))) float v4f;
typedef __attribute__((ext_vector_type(8))) float v8f;

#define B_   8
#define N_   16384
#define M_   4096
#define KNN  32
#define WPB  4            // waves per block (wave32); each takes N_/WPB points
#define SEG  (N_ / WPB)   // 4096 points per wave-segment
#define SSTR 36           // LDS row stride (floats): 16B-aligned b128, low bank conflicts
#define WS   (32 * SSTR)  // per-wave LDS region stride (elements)

// Block = 4 waves sharing 32 queries; wave w scans points [w*SEG, (w+1)*SEG).
// Score(m, j) = -2*q_m . x_j + ||x_j||^2 (ordering-equivalent to squared distance),
// computed 32x32 per iteration with four V_WMMA_F32_16X16X4_F32 (K = 4 exactly):
//   A row = (-2qx, -2qy, -2qz, 1),  B col = (x, y, z, ||x||^2)
// Per-wave top-32 kept in LDS; waves sort their lists, then wave 0 4-way merges.
__global__ __launch_bounds__(WPB * 32)
void knn_topk_wmma_kernel(const float* __restrict__ xyz,
                          const float* __restrict__ nxyz,
                          int* __restrict__ out)
{
    __shared__ float s_sc[WPB][WS];     // score tile during loop; sorted values after
    __shared__ float s_tv[WPB][WS];     // per-lane top-32 values (unsorted)
    __shared__ int   s_ti[WPB][WS];     // per-lane top-32 indices (unsorted)
    __shared__ int   s_si[WPB][WS];     // per-lane sorted indices
    __shared__ float s_qa[WPB][32 * 4]; // packed query rows

    const int tid = threadIdx.x;
    const int w   = tid >> 5;
    const int l   = tid & 31;
    const int m   = l & 15;
    const bool hi = l >= 16;

    const int b     = blockIdx.x >> 7;          // M_/32 = 128 query-groups per batch
    const int qbase = (blockIdx.x & 127) << 5;

    // ---- stage packed query rows; build persistent A fragments ------------
    {
        const float* qp = nxyz + ((size_t)b * M_ + qbase + l) * 3;
        float qx = qp[0], qy = qp[1], qz = qp[2];
        float* qa = &s_qa[w][l * 4];
        qa[0] = -2.0f * qx; qa[1] = -2.0f * qy; qa[2] = -2.0f * qz; qa[3] = 1.0f;
    }
    // A 16x4 f32 layout: lanes 0-15 hold (K0,K1) of row M=lane, lanes 16-31 hold (K2,K3)
    const int koff = hi ? 2 : 0;
    v2f a0 = *(const v2f*)&s_qa[w][ m       * 4 + koff];  // queries qbase+0..15
    v2f a1 = *(const v2f*)&s_qa[w][(16 + m) * 4 + koff];  // queries qbase+16..31

    // ---- init per-lane top-k state ----------------------------------------
    const float INF = __builtin_huge_valf();
    {
        float* tv = &s_tv[w][l * SSTR];
        int*   ti = &s_ti[w][l * SSTR];
        #pragma unroll
        for (int k = 0; k < KNN; ++k) { tv[k] = INF; ti[k] = 0; }
    }
    float wmax  = INF;   // current worst of the kept 32
    int   wslot = 0;     // its slot

    const float* xb    = xyz + (size_t)b * N_ * 3;
    const int    jbase = w * SEG;

    for (int j0 = jbase; j0 < jbase + SEG; j0 += 32) {
        // B 4x16 f32 fragments: lanes 0-15 supply (x,y), lanes 16-31 supply (z, ||x||^2)
        const float* p0 = xb + (size_t)(j0 + m) * 3;
        float x0 = p0[0], y0 = p0[1], z0 = p0[2];
        float n0 = __fmaf_rn(x0, x0, __fmaf_rn(y0, y0, z0 * z0));
        const float* p1 = p0 + 48;                       // +16 points
        float x1 = p1[0], y1 = p1[1], z1 = p1[2];
        float n1 = __fmaf_rn(x1, x1, __fmaf_rn(y1, y1, z1 * z1));
        if (j0 + 32 < jbase + SEG) __builtin_prefetch(p0 + 96, 0, 1);  // global_prefetch_b8

        v2f bt0, bt1;
        bt0.x = hi ? z0 : x0;  bt0.y = hi ? n0 : y0;
        bt1.x = hi ? z1 : x1;  bt1.y = hi ? n1 : y1;

        v8f c0 = {0.f,0.f,0.f,0.f,0.f,0.f,0.f,0.f};
        v8f d00 = __builtin_amdgcn_wmma_f32_16x16x4_f32(false, a0, false, bt0, (short)0, c0, false, false);
        v8f d01 = __builtin_amdgcn_wmma_f32_16x16x4_f32(false, a0, false, bt1, (short)0, c0, false, false);
        v8f d10 = __builtin_amdgcn_wmma_f32_16x16x4_f32(false, a1, false, bt0, (short)0, c0, false, false);
        v8f d11 = __builtin_amdgcn_wmma_f32_16x16x4_f32(false, a1, false, bt1, (short)0, c0, false, false);

        // scatter D tiles to LDS: row = wave-local query 0..31, col = chunk point 0..31
        {
            const int r0 = hi ? 8 : 0;
            float* sc = s_sc[w];
            #pragma unroll
            for (int r = 0; r < 8; ++r) {
                sc[(r0 + r     ) * SSTR + m     ] = d00[r];
                sc[(r0 + r     ) * SSTR + 16 + m] = d01[r];
                sc[(r0 + r + 16) * SSTR + m     ] = d10[r];
                sc[(r0 + r + 16) * SSTR + 16 + m] = d11[r];
            }
        }

        // each lane scans its own query's 32 candidates; min4 pre-filter keeps
        // the common case at one branch per ds_load_b128
        const float* srow = &s_sc[w][l * SSTR];
        #pragma unroll 1
        for (int t = 0; t < 32; t += 4) {
            v4f sv = *(const v4f*)(srow + t);            // ds_load_b128
            float m01 = fminf(sv[0], sv[1]);
            float m23 = fminf(sv[2], sv[3]);
            if (fminf(m01, m23) < wmax) {                // rare path
                #pragma unroll
                for (int u = 0; u < 4; ++u) {
                    float s = sv[u];
                    if (s < wmax) {
                        s_tv[w][l * SSTR + wslot] = s;
                        s_ti[w][l * SSTR + wslot] = j0 + t + u;
                        float nm = -INF; int ns = 0;     // rescan for new worst
                        #pragma unroll 1
                        for (int k = 0; k < 8; ++k) {
                            v4f v = *(const v4f*)&s_tv[w][l * SSTR + 4 * k];
                            #pragma unroll
                            for (int c = 0; c < 4; ++c) {
                                bool gtr = v[c] > nm;
                                nm = gtr ? v[c] : nm;
                                ns = gtr ? 4 * k + c : ns;
                            }
                        }
                        wmax = nm; wslot = ns;
                    }
                }
            }
        }
    }

    // ---- each wave sorts its own list (ascending) into s_sc / s_si --------
    #pragma unroll 1
    for (int slot = 0; slot < KNN; ++slot) {
        float bm = INF; int bs = 0;
        #pragma unroll 1
        for (int k = 0; k < 8; ++k) {
            v4f v = *(const v4f*)&s_tv[w][l * SSTR + 4 * k];
            #pragma unroll
            for (int c = 0; c < 4; ++c) {
                bool lt = v[c] < bm;
                bm = lt ? v[c] : bm;
                bs = lt ? 4 * k + c : bs;
            }
        }
        s_sc[w][l * SSTR + slot] = bm;
        s_si[w][l * SSTR + slot] = s_ti[w][l * SSTR + bs];
        s_tv[w][l * SSTR + bs]  = INF;                   // remove and continue
    }
    // sentinel pad so the merge can read one past a drained head
    #pragma unroll
    for (int k = 0; k < 4; ++k) s_sc[w][l * SSTR + KNN + k] = INF;

    __syncthreads();

    // ---- wave 0: 4-way merge of sorted segment lists, emit ascending ------
    if (w == 0) {
        const float* scf = &s_sc[0][0];
        const int*   sif = &s_si[0][0];
        const int    base = l * SSTR;
        float v0 = scf[0 * WS + base], v1 = scf[1 * WS + base];
        float v2 = scf[2 * WS + base], v3 = scf[3 * WS + base];
        int h0 = 0, h1 = 0, h2 = 0, h3 = 0;
        int* op = out + (((size_t)b * M_ + qbase + l) * KNN);
        #pragma unroll 1
        for (int slot = 0; slot < KNN; ++slot) {
            float mv = v0; int sel = 0;                  // strict < : ties pick lower segment
            if (v1 < mv) { mv = v1; sel = 1; }
            if (v2 < mv) { mv = v2; sel = 2; }
            if (v3 < mv) { mv = v3; sel = 3; }
            int hsel = sel == 0 ? h0 : sel == 1 ? h1 : sel == 2 ? h2 : h3;
            int addr = sel * WS + base + hsel;
            op[slot] = sif[addr];
            float nv = scf[addr + 1];                    // sentinel-safe
            if      (sel == 0) { h0 = hsel + 1; v0 = nv; }
            else if (sel == 1) { h1 = hsel + 1; v1 = nv; }
            else if (sel == 2) { h2 = hsel + 1; v2 = nv; }
            else               { h3 = hsel + 1; v3 = nv; }
        }
    }
}

extern "C" void kernel_launch(void* const* d_in, const int* in_sizes, int n_in,
                              void* d_out, int out_size, void* d_ws, size_t ws_size,
                              hipStream_t stream) {
    (void)in_sizes; (void)n_in; (void)d_ws; (void)ws_size; (void)out_size;
    const float* xyz  = (const float*)d_in[0];   // (8, 16384, 3)
    const float* nxyz = (const float*)d_in[1];   // (8, 4096, 3)
    int* out = (int*)d_out;                      // (8, 4096, 32) int32

    dim3 block(WPB * 32);                        // 128 threads = 4 wave32
    dim3 grid(B_ * (M_ / 32));                   // 1024 blocks, one per 32-query group
    hipLaunchKernelGGL(knn_topk_wmma_kernel, grid, block, 0, stream, xyz, nxyz, out);
}